// FocalLoss_36670430773655
// MI455X (gfx1250) — compile-verified
//
#include <hip/hip_runtime.h>
#include <stdint.h>

#define ALPHA_F 0.25f
#define A_N 120000
#define B_N 8
#define C_N 80
#define M_N 32
#define TB  128   // anchors per block == threads per block

typedef uint32_t u32x4 __attribute__((ext_vector_type(4)));
typedef uint32_t u32x8 __attribute__((ext_vector_type(8)));

// ws layout (floats): [0] anchor-max bits, [1..8] cls_sum, [9..16] giou_sum, [17..24] npos
__global__ void zero_ws_kernel(float* ws) {
    int i = threadIdx.x;
    if (i < 25) ws[i] = 0.0f;
}

__global__ void anchor_max_kernel(const float* __restrict__ anc, unsigned* mbits, int n) {
    float lm = 0.0f;  // true max is positive (y2 up to ~928), so 0 init is safe
    for (int i = blockIdx.x * blockDim.x + threadIdx.x; i < n; i += gridDim.x * blockDim.x)
        lm = fmaxf(lm, anc[i]);
    for (int off = 16; off > 0; off >>= 1)
        lm = fmaxf(lm, __shfl_xor(lm, off, 32));
    if ((threadIdx.x & 31) == 0)
        atomicMax(mbits, __float_as_uint(lm));  // nonneg floats: uint order == float order
}

__global__ __launch_bounds__(TB) void focal_main_kernel(
    const float* __restrict__ cls, const float* __restrict__ reg,
    const float* __restrict__ anc, const float* __restrict__ ann,
    const unsigned* __restrict__ mbits, float* __restrict__ acc)
{
    __shared__ float tile[TB * C_N];    // 40 KB classification tile (TDM target)
    __shared__ float annRaw[M_N * 5];
    __shared__ float annN[M_N * 4];
    __shared__ float annArea[M_N];
    __shared__ int   sCase[TB];         // 0=ignore band, 1=negative, 2=positive
    __shared__ int   sCls[TB];
    __shared__ float sRed[3];           // [0] cls loss, [1] giou loss, [2] npos

    const int b   = blockIdx.y;
    const int a0  = blockIdx.x * TB;
    const int nA  = min(TB, A_N - a0);
    const int nE  = nA * C_N;           // tile elements (<= 10240)
    const int tid = threadIdx.x;

    // ---- Wave 0: issue async Tensor-DMA of the contiguous cls tile into LDS.
    //      Stream is 307 MB read-once (> 192 MB L2): use non-temporal hint so it
    //      doesn't thrash L2 lines holding reused anchors/accumulators. ----
    if ((tid >> 5) == 0) {
        uint64_t ga  = (uint64_t)(uintptr_t)(cls + ((size_t)b * A_N + a0) * C_N);
        uint32_t lds = (uint32_t)(uintptr_t)&tile[0];   // flat LDS addr: low 32 bits = LDS offset
        u32x4 d0;
        d0.x = 1u;                                              // count=1, user descriptor
        d0.y = lds;                                             // lds_addr
        d0.z = (uint32_t)ga;                                    // global_addr[31:0]
        d0.w = ((uint32_t)(ga >> 32) & 0x01FFFFFFu) | 0x80000000u; // addr[56:32], type=2
        u32x8 d1;
        d1.s0 = 2u << 16;                                       // workgroup_mask=0, data_size=4B
        d1.s1 = ((uint32_t)nE & 0xFFFFu) << 16;                 // tensor_dim0[15:0]
        d1.s2 = ((uint32_t)nE >> 16) | (1u << 16);              // tensor_dim0[31:16], tensor_dim1=1
        d1.s3 = ((uint32_t)nE & 0xFFFFu) << 16;                 // tensor_dim1 hi=0, tile_dim0=nE
        d1.s4 = 1u;                                             // tile_dim1=1, tile_dim2=0
        d1.s5 = (uint32_t)nE;                                   // tensor_dim0_stride[31:0]
        d1.s6 = 0u;
        d1.s7 = 0u;
        asm volatile("tensor_load_to_lds %0, %1 th:TH_LOAD_NT scope:SCOPE_DEV"
                     :: "s"(d0), "s"(d1) : "memory");
    }

    // ---- Stage annotations, init block reductions ----
    for (int i = tid; i < M_N * 5; i += TB) annRaw[i] = ann[(size_t)b * M_N * 5 + i];
    if (tid == 0) { sRed[0] = 0.f; sRed[1] = 0.f; sRed[2] = 0.f; }
    __syncthreads();

    const float mval = __uint_as_float(mbits[0]);
    if (tid < M_N) {
        float x1 = annRaw[tid * 5 + 0] / mval;
        float y1 = annRaw[tid * 5 + 1] / mval;
        float x2 = annRaw[tid * 5 + 2] / mval;
        float y2 = annRaw[tid * 5 + 3] / mval;
        annN[tid * 4 + 0] = x1; annN[tid * 4 + 1] = y1;
        annN[tid * 4 + 2] = x2; annN[tid * 4 + 3] = y2;
        annArea[tid] = (x2 - x1) * (y2 - y1);
    }
    __syncthreads();

    // ---- Phase A: IoU / argmax over 32 GT boxes + GIoU for positives ----
    if (tid < nA) {
        const int a = a0 + tid;
        const float4 av = ((const float4*)anc)[a];   // [y1, x1, y2, x2]
        const float ay1 = av.x, ax1 = av.y, ay2 = av.z, ax2 = av.w;
        const float ay1n = ay1 / mval, ax1n = ax1 / mval;
        const float ay2n = ay2 / mval, ax2n = ax2 / mval;
        const float areaA = (ay2n - ay1n) * (ax2n - ax1n);
        float best = -1.0f; int bj = 0;
        #pragma unroll 4
        for (int j = 0; j < M_N; ++j) {
            float bx1 = annN[4*j], by1 = annN[4*j+1], bx2 = annN[4*j+2], by2 = annN[4*j+3];
            float iw = fmaxf(fminf(ax2n, bx2) - fmaxf(ax1n, bx1), 0.0f);
            float ih = fmaxf(fminf(ay2n, by2) - fmaxf(ay1n, by1), 0.0f);
            float ua = fmaxf(areaA + annArea[j] - iw * ih, 1e-8f);
            float iou = iw * ih / ua;
            if (iou > best) { best = iou; bj = j; }   // first-max == jnp.argmax
        }
        const int cse = (best >= 0.5f) ? 2 : ((best < 0.4f) ? 1 : 0);
        sCase[tid] = cse;
        sCls[tid]  = (int)annRaw[bj * 5 + 4];

        if (cse == 2) {  // regression GIoU, reference epsilon/clamp sequence
            const float4 rv = ((const float4*)reg)[(size_t)b * A_N + a]; // ty,tx,th,tw
            float aw = ax2 - ax1, ah = ay2 - ay1;
            float acx = ax1 + 0.5f * aw, acy = ay1 + 0.5f * ah;
            float pcx = rv.y * aw + acx, pcy = rv.x * ah + acy;
            float pw = expf(rv.w) * aw,  ph  = expf(rv.z) * ah;
            float p0 = fmaxf(pcx - 0.5f * pw, 0.f), p1 = fmaxf(pcy - 0.5f * ph, 0.f);
            float p2 = fmaxf(pcx + 0.5f * pw, 0.f), p3 = fmaxf(pcy + 0.5f * ph, 0.f);
            float q0 = fmaxf(annRaw[bj*5+0], 0.f), q1 = fmaxf(annRaw[bj*5+1], 0.f);
            float q2 = fmaxf(annRaw[bj*5+2], 0.f), q3 = fmaxf(annRaw[bj*5+3], 0.f);
            float px1 = fminf(p0,p2), py1 = fminf(p1,p3), px2 = fmaxf(p0,p2), py2 = fmaxf(p1,p3);
            float gx1 = fminf(q0,q2), gy1 = fminf(q1,q3), gx2 = fmaxf(q0,q2), gy2 = fmaxf(q1,q3);
            float iw = fmaxf(fminf(px2,gx2) - fmaxf(px1,gx1), 0.f);
            float ih = fmaxf(fminf(py2,gy2) - fmaxf(py1,gy1), 0.f);
            float inter = iw * ih;
            float areaP = fmaxf((px2-px1)*(py2-py1), 1e-6f);
            float areaG = fmaxf((gx2-gx1)*(gy2-gy1), 1e-6f);
            float uni   = areaP + areaG - inter;
            float iou   = inter / (uni + 1e-7f);
            float wc = fmaxf(fmaxf(px2,gx2) - fminf(px1,gx1), 1e-6f);
            float hc = fmaxf(fmaxf(py2,gy2) - fminf(py1,gy1), 1e-6f);
            float areaC = wc * hc;
            float giou = iou - (areaC - uni) / (areaC + 1e-7f);
            giou = fminf(fmaxf(giou, -1.f), 1.f);
            atomicAdd(&sRed[1], 1.0f - giou);
            atomicAdd(&sRed[2], 1.0f);
        }
    }

    // ---- Wait for TDM tile, then barrier makes it + meta visible to all waves ----
    if ((tid >> 5) == 0) __builtin_amdgcn_s_wait_tensorcnt(0);
    __syncthreads();
    if (tid == 0) {
        atomicAdd(&acc[9  + b], sRed[1]);
        atomicAdd(&acc[17 + b], sRed[2]);
    }

    // ---- Phase B: focal loss streamed from LDS (float4; 80 % 4 == 0) ----
    float lsum = 0.0f;
    const int nF4 = nE >> 2;
    for (int i = tid; i < nF4; i += TB) {
        int anch = i / 20;                  // 20 float4 per anchor row
        int cse = sCase[anch];
        if (cse == 0) continue;             // [0.4,0.5) band: zero loss
        float4 v = ((const float4*)tile)[i];
        int cbase = (i - anch * 20) * 4;
        int acls  = (cse == 2) ? sCls[anch] : -1;
        float cv[4] = {v.x, v.y, v.z, v.w};
        #pragma unroll
        for (int k = 0; k < 4; ++k) {
            float c = fminf(fmaxf(cv[k], 5e-4f), 1.0f - 5e-4f);
            if (cbase + k == acls) {
                float u = 1.0f - c;
                lsum += ALPHA_F * u * u * (-logf(c));          // t==1 term
            } else {
                lsum += (1.0f - ALPHA_F) * c * c * (-logf(1.0f - c)); // t==0 term
            }
        }
    }
    atomicAdd(&sRed[0], lsum);
    __syncthreads();
    if (tid == 0) atomicAdd(&acc[1 + b], sRed[0]);
}

__global__ void finalize_kernel(const float* __restrict__ acc, float* __restrict__ out) {
    if (threadIdx.x == 0 && blockIdx.x == 0) {
        float cs = 0.f, rs = 0.f;
        for (int b = 0; b < B_N; ++b) {
            float np = acc[17 + b];
            float d  = fmaxf(np, 1.0f);
            cs += acc[1 + b] / d;
            rs += (np > 0.f) ? (acc[9 + b] / d) : 0.f;
        }
        cs *= (1.0f / B_N);
        rs *= (1.0f / B_N);
        out[0] = cs + rs;
        out[1] = cs;
        out[2] = rs;
    }
}

extern "C" void kernel_launch(void* const* d_in, const int* in_sizes, int n_in,
                              void* d_out, int out_size, void* d_ws, size_t ws_size,
                              hipStream_t stream) {
    const float* cls = (const float*)d_in[0];   // (B, A, C)
    const float* reg = (const float*)d_in[1];   // (B, A, 4)
    const float* anc = (const float*)d_in[2];   // (1, A, 4)
    const float* ann = (const float*)d_in[3];   // (B, M, 5)
    float* ws  = (float*)d_ws;
    float* out = (float*)d_out;

    zero_ws_kernel<<<1, 32, 0, stream>>>(ws);
    anchor_max_kernel<<<256, 256, 0, stream>>>(anc, (unsigned*)ws, A_N * 4);
    dim3 grid((A_N + TB - 1) / TB, B_N);
    focal_main_kernel<<<grid, TB, 0, stream>>>(cls, reg, anc, ann,
                                               (const unsigned*)ws, ws);
    finalize_kernel<<<1, 32, 0, stream>>>(ws, out);
}